// KNN_CHECK_68582037783103
// MI455X (gfx1250) — compile-verified
//
#include <hip/hip_runtime.h>

// ---------------- problem constants (from reference setup_inputs) -----------
namespace {
constexpr int B      = 8;
constexpr int N      = 8192;
constexpr int S      = 1024;
constexpr int K      = 32;
constexpr int TOPC   = 5;
constexpr int MAXIT  = 16;
constexpr int QT     = 16;   // queries per workgroup (one WMMA N-tile)
constexpr int WAVES  = 4;    // waves per workgroup, splitting the N dimension
constexpr int NPW    = N / WAVES;  // points scanned per wave

// workspace layout, in 4-byte words
constexpr size_t OFF_CURP = 0;                            // B*S*3 float
constexpr size_t OFF_OUTP = OFF_CURP + (size_t)B * S * 3; // B*S*3 float
constexpr size_t OFF_CURI = OFF_OUTP + (size_t)B * S * 3; // B*S   int
constexpr size_t OFF_CONV = OFF_CURI + (size_t)B * S;     // B*S   int
constexpr size_t OFF_ACT  = OFF_CONV + (size_t)B * S;     // B*S   int
constexpr size_t OFF_OUTI = OFF_ACT  + (size_t)B * S;     // B*S   int
constexpr size_t OFF_IND  = OFF_OUTI + (size_t)B * S;     // B*S*K int
constexpr size_t OFF_OUTG = OFF_IND  + (size_t)B * S * K; // B*S*K int
} // namespace

typedef __attribute__((ext_vector_type(2))) float v2f;
typedef __attribute__((ext_vector_type(8))) float v8f;

// ---------------- init: cur state from inputs -------------------------------
__global__ void init_k(const float* __restrict__ center,
                       const int* __restrict__ idx0,
                       float* __restrict__ wsf) {
  int tid = blockIdx.x * blockDim.x + threadIdx.x;
  if (tid >= B * S) return;
  int* wsi = (int*)wsf;
  wsf[OFF_CURP + (size_t)tid * 3 + 0] = center[(size_t)tid * 3 + 0];
  wsf[OFF_CURP + (size_t)tid * 3 + 1] = center[(size_t)tid * 3 + 1];
  wsf[OFF_CURP + (size_t)tid * 3 + 2] = center[(size_t)tid * 3 + 2];
  wsi[OFF_CURI + tid] = idx0[tid];
  wsi[OFF_CONV + tid] = MAXIT;
  wsi[OFF_ACT  + tid] = 1;
}

// ---------------- KNN: WMMA-scored top-32 -----------------------------------
// 4 waves per workgroup, 16 queries per workgroup; each wave scans N/4 points.
// Score = ||p||^2 - 2 q.p  (same ordering as squared distance; ||q||^2 is a
// per-row constant). One V_WMMA_F32_16X16X4_F32 per 16-point chunk:
//   A (16 points x 4)  : lane m: {px,py} ; lane m+16: {pz, ||p||^2}
//   B (4 x 16 queries) : lane n: {-2qx,-2qy} ; lane n+16: {-2qz, 1}
//   D (16x16 f32)      : VGPR v, lane n -> point v, query n  (v+8 for n+16)
// Selection: per-lane top-32 in LDS -> per-wave sorted 32 -> 4-way merge.
__global__ __launch_bounds__(WAVES * 32) void knn_k(const float* __restrict__ xyz,
                                                    float* __restrict__ wsf) {
  const int wg    = blockIdx.x;
  const int b     = wg / (S / QT);
  const int qt    = wg % (S / QT);
  const int w     = threadIdx.x >> 5;   // wave id 0..3
  const int lane  = threadIdx.x & 31;   // 0..31
  const int qlane = lane & 15;
  const int half  = lane >> 4;
  const int q     = qt * QT + qlane;
  const int nbase = w * NPW;            // this wave's slice of the cloud

  const float* pc = xyz + (size_t)b * N * 3;
  int* wsi = (int*)wsf;
  const float INF = __builtin_inff();

  // query -> B operand (loop invariant, branchless)
  const float* cp = wsf + OFF_CURP + ((size_t)b * S + q) * 3;
  const float qx = cp[0], qy = cp[1], qz = cp[2];
  v2f bq;
  bq.x = half ? (-2.0f * qz) : (-2.0f * qx);
  bq.y = half ? 1.0f         : (-2.0f * qy);

  // per-(wave,lane) top-32 candidate lists, lane-major (lane==bank)
  __shared__ float ldsD[WAVES * 32 * 32];
  __shared__ int   ldsI[WAVES * 32 * 32];
  // per-wave sorted top-32 per query
  __shared__ float sortD[WAVES * K * QT];
  __shared__ int   sortI[WAVES * K * QT];

  float* Dw = ldsD + w * 32 * 32;
  int*   Iw = ldsI + w * 32 * 32;
  for (int s2 = 0; s2 < 32; ++s2) {
    Dw[s2 * 32 + lane] = INF;
    Iw[s2 * 32 + lane] = 0x7fffffff;
  }
  float curMax  = INF;
  int   curSlot = 0;

  // ---- phase 1: scan this wave's 2048 points, 16 per WMMA ----
#pragma unroll 2
  for (int base = 0; base < NPW; base += 16) {
    const float* pp = pc + (size_t)(nbase + base + qlane) * 3;
    __builtin_prefetch((const void*)(pp + 48), 0, 3);   // next 16-point chunk
    const float px = pp[0], py = pp[1], pz = pp[2];
    const float pn = px * px + py * py + pz * pz;
    v2f a;
    a.x = half ? pz : px;       // branchless lane-select (v_cndmask)
    a.y = half ? pn : py;
    v8f c = {};
    v8f d = __builtin_amdgcn_wmma_f32_16x16x4_f32(
        /*neg_a=*/false, a, /*neg_b=*/false, bq,
        /*c_mod=*/(short)0, c, /*reuse_a=*/false, /*reuse_b=*/false);

    const int pbase = nbase + base + half * 8;
#pragma unroll
    for (int v = 0; v < 8; ++v) {
      const float sc   = d[v];
      const int   pidx = pbase + v;
      if (sc < curMax) {
        Dw[curSlot * 32 + lane] = sc;
        Iw[curSlot * 32 + lane] = pidx;
        float m = -INF; int ms = 0;
        for (int s2 = 0; s2 < 32; ++s2) {
          const float t = Dw[s2 * 32 + lane];
          if (t > m) { m = t; ms = s2; }
        }
        curMax = m; curSlot = ms;
      }
    }
  }
  __syncthreads();

  // ---- phase 2: per wave, 64 -> sorted 32 per query (ties -> lower idx) ----
  if (half == 0) {
    unsigned long long used = 0ull;
    for (int j = 0; j < K; ++j) {
      float bd = INF; int bi = 0x7fffffff; int bt = 0;
      for (int t = 0; t < 64; ++t) {
        if ((used >> t) & 1ull) continue;
        const int ln = qlane + ((t >> 5) << 4);
        const int sl = t & 31;
        const float dv = Dw[sl * 32 + ln];
        const int   iv = Iw[sl * 32 + ln];
        if (dv < bd || (dv == bd && iv < bi)) { bd = dv; bi = iv; bt = t; }
      }
      used |= (1ull << bt);
      sortD[(w * K + j) * QT + qlane] = bd;
      sortI[(w * K + j) * QT + qlane] = bi;
    }
  }
  __syncthreads();

  // ---- phase 3: 4-way merge of sorted lists (wave 0, one lane per query) ---
  if (threadIdx.x < QT) {
    const int qq = threadIdx.x;
    int h0 = 0, h1 = 0, h2 = 0, h3 = 0;
    int* out = wsi + OFF_IND + ((size_t)b * S + qt * QT + qq) * K;
    for (int j = 0; j < K; ++j) {
      const float d0 = (h0 < K) ? sortD[(0 * K + h0) * QT + qq] : INF;
      const float d1 = (h1 < K) ? sortD[(1 * K + h1) * QT + qq] : INF;
      const float d2 = (h2 < K) ? sortD[(2 * K + h2) * QT + qq] : INF;
      const float d3 = (h3 < K) ? sortD[(3 * K + h3) * QT + qq] : INF;
      const int   i0 = (h0 < K) ? sortI[(0 * K + h0) * QT + qq] : 0x7fffffff;
      const int   i1 = (h1 < K) ? sortI[(1 * K + h1) * QT + qq] : 0x7fffffff;
      const int   i2 = (h2 < K) ? sortI[(2 * K + h2) * QT + qq] : 0x7fffffff;
      const int   i3 = (h3 < K) ? sortI[(3 * K + h3) * QT + qq] : 0x7fffffff;
      float bd = d0; int bi = i0; int sel = 0;
      if (d1 < bd || (d1 == bd && i1 < bi)) { bd = d1; bi = i1; sel = 1; }
      if (d2 < bd || (d2 == bd && i2 < bi)) { bd = d2; bi = i2; sel = 2; }
      if (d3 < bd || (d3 == bd && i3 < bi)) { bd = d3; bi = i3; sel = 3; }
      out[j] = bi;
      if      (sel == 0) ++h0;
      else if (sel == 1) ++h1;
      else if (sel == 2) ++h2;
      else               ++h3;
    }
  }
}

// ---------------- per-iteration centroid check + move -----------------------
__global__ void update_k(const float* __restrict__ xyz,
                         float* __restrict__ wsf, int it) {
  int tid = blockIdx.x * blockDim.x + threadIdx.x;
  if (tid >= B * S) return;
  int* wsi = (int*)wsf;
  if (!wsi[OFF_ACT + tid]) return;          // masked out: state untouched

  const int b = tid / S;
  const float* pc = xyz + (size_t)b * N * 3;
  const int* ind = wsi + OFF_IND + (size_t)tid * K;

  float gx = 0.f, gy = 0.f, gz = 0.f;
  for (int k2 = 0; k2 < K; ++k2) {
    const float* pp = pc + (size_t)ind[k2] * 3;
    gx += pp[0]; gy += pp[1]; gz += pp[2];
  }
  const float inv = 1.0f / (float)K;
  gx *= inv; gy *= inv; gz *= inv;

  float d0 = 0.f, best = __builtin_inff();
  int cnt = 0, bestk = 0;
  for (int k2 = 0; k2 < K; ++k2) {
    const float* pp = pc + (size_t)ind[k2] * 3;
    const float dx = pp[0] - gx, dy = pp[1] - gy, dz = pp[2] - gz;
    const float d2 = dx * dx + dy * dy + dz * dz;
    if (k2 == 0) d0 = d2; else cnt += (d2 < d0) ? 1 : 0;
    if (d2 < best) { best = d2; bestk = k2; }
  }
  const bool in_top = (cnt < TOPC);         // neighbor 0 within 5 closest to gc

  if (in_top) {                             // converged this iteration
    wsi[OFF_CONV + tid] = it;
    wsf[OFF_OUTP + (size_t)tid * 3 + 0] = wsf[OFF_CURP + (size_t)tid * 3 + 0];
    wsf[OFF_OUTP + (size_t)tid * 3 + 1] = wsf[OFF_CURP + (size_t)tid * 3 + 1];
    wsf[OFF_OUTP + (size_t)tid * 3 + 2] = wsf[OFF_CURP + (size_t)tid * 3 + 2];
    wsi[OFF_OUTI + tid] = wsi[OFF_CURI + tid];
    for (int k2 = 0; k2 < K; ++k2)
      wsi[OFF_OUTG + (size_t)tid * K + k2] = ind[k2];
    wsi[OFF_ACT + tid] = 0;
  } else {                                  // move to neighbor closest to gc
    const int ni = ind[bestk];
    wsi[OFF_CURI + tid] = ni;
    const float* pp = pc + (size_t)ni * 3;
    wsf[OFF_CURP + (size_t)tid * 3 + 0] = pp[0];
    wsf[OFF_CURP + (size_t)tid * 3 + 1] = pp[1];
    wsf[OFF_CURP + (size_t)tid * 3 + 2] = pp[2];
  }
}

// ---------------- finalize never-converged queries --------------------------
__global__ void final_k(float* __restrict__ wsf) {
  int tid = blockIdx.x * blockDim.x + threadIdx.x;
  if (tid >= B * S) return;
  int* wsi = (int*)wsf;
  if (!wsi[OFF_ACT + tid]) return;
  wsf[OFF_OUTP + (size_t)tid * 3 + 0] = wsf[OFF_CURP + (size_t)tid * 3 + 0];
  wsf[OFF_OUTP + (size_t)tid * 3 + 1] = wsf[OFF_CURP + (size_t)tid * 3 + 1];
  wsf[OFF_OUTP + (size_t)tid * 3 + 2] = wsf[OFF_CURP + (size_t)tid * 3 + 2];
  wsi[OFF_OUTI + tid] = wsi[OFF_CURI + tid];
  for (int k2 = 0; k2 < K; ++k2)
    wsi[OFF_OUTG + (size_t)tid * K + k2] = wsi[OFF_IND + (size_t)tid * K + k2];
}

// ---------------- stable (conv_it, index) ordering + scatter to d_out -------
__global__ void emit_k(const float* __restrict__ wsf, float* __restrict__ outf) {
  int tid = blockIdx.x * blockDim.x + threadIdx.x;
  if (tid >= B * S) return;
  const int* wsi = (const int*)wsf;
  const int b = tid / S, s = tid % S;

  const int key = wsi[OFF_CONV + tid] * (S + 1) + s;    // all keys distinct
  const int* convb = wsi + OFF_CONV + (size_t)b * S;
  int rank = 0;
  for (int t = 0; t < S; ++t)
    rank += (convb[t] * (S + 1) + t < key) ? 1 : 0;

  const size_t dst = (size_t)b * S + rank;
  outf[dst * 3 + 0] = wsf[OFF_OUTP + (size_t)tid * 3 + 0];
  outf[dst * 3 + 1] = wsf[OFF_OUTP + (size_t)tid * 3 + 1];
  outf[dst * 3 + 2] = wsf[OFF_OUTP + (size_t)tid * 3 + 2];
  int* outi = (int*)(outf + (size_t)B * S * 3);
  outi[dst] = wsi[OFF_OUTI + tid];
  int* outg = outi + (size_t)B * S;
  for (int k2 = 0; k2 < K; ++k2)
    outg[dst * K + k2] = wsi[OFF_OUTG + (size_t)tid * K + k2];
}

// ---------------- driver ----------------------------------------------------
extern "C" void kernel_launch(void* const* d_in, const int* in_sizes, int n_in,
                              void* d_out, int out_size, void* d_ws, size_t ws_size,
                              hipStream_t stream) {
  (void)in_sizes; (void)n_in; (void)out_size; (void)ws_size;
  const float* xyz    = (const float*)d_in[0];
  const float* center = (const float*)d_in[1];
  const int*   idx0   = (const int*)d_in[2];
  float* wsf  = (float*)d_ws;
  float* outf = (float*)d_out;

  const int nthreads = B * S;
  const int nblk = (nthreads + 255) / 256;
  const int knn_blocks = B * (S / QT);

  init_k<<<nblk, 256, 0, stream>>>(center, idx0, wsf);
  for (int it = 0; it < MAXIT; ++it) {
    knn_k<<<knn_blocks, WAVES * 32, 0, stream>>>(xyz, wsf);
    update_k<<<nblk, 256, 0, stream>>>(xyz, wsf, it);
  }
  knn_k<<<knn_blocks, WAVES * 32, 0, stream>>>(xyz, wsf);
  final_k<<<nblk, 256, 0, stream>>>(wsf);
  emit_k<<<nblk, 256, 0, stream>>>(wsf, outf);
}